// TextureExtractor_13048110645620
// MI455X (gfx1250) — compile-verified
//
#include <hip/hip_runtime.h>

#define HH 320
#define WW 320
#define NPIX (HH * WW)
#define RAD 15
#define DD 31

typedef float v2f __attribute__((ext_vector_type(2)));
typedef float v8f __attribute__((ext_vector_type(8)));

__device__ __forceinline__ int refl(int p, int n) {
  // reflect-101 (cv2 / jnp.pad "reflect"): -1 -> 1, n -> n-2
  p = (p < 0) ? -p : p;
  p = (p >= n) ? (2 * n - 2 - p) : p;
  return p;
}

// ---------------------------------------------------------------- gray + init
__global__ __launch_bounds__(256) void k_gray(const float* __restrict__ img,
                                              float* __restrict__ gray,
                                              unsigned* __restrict__ maxb) {
  int i = blockIdx.x * blockDim.x + threadIdx.x;
  if (i == 0) *maxb = 0u;  // ws is poisoned before timing; re-init every call
  if (i < NPIX) {
    gray[i] = 0.299f * img[i] + 0.587f * img[NPIX + i] + 0.114f * img[2 * NPIX + i];
  }
}

// ------------------------------------------------------- weight/BN prep
// A-fragment layout for V_WMMA_F32_16X16X4_F32 (wave32):
//   lane l (l<16):  VGPR0 = A[M=l][K=0],  VGPR1 = A[M=l][K=1]
//   lane l (l>=16): VGPR0 = A[M=l-16][K=2], VGPR1 = A[M=l-16][K=3]
// Pre-pack weights so each lane loads its two elements as one float2, with
// consecutive lanes contiguous (coalesced 256B/wave global_load_b64).
__global__ __launch_bounds__(256) void k_prep(
    const float* __restrict__ w2, const float* __restrict__ b2,
    const float* __restrict__ g2, const float* __restrict__ be2,
    const float* __restrict__ m2, const float* __restrict__ v2_,
    const float* __restrict__ b1, const float* __restrict__ g1,
    const float* __restrict__ be1, const float* __restrict__ m1,
    const float* __restrict__ v1_,
    float2* __restrict__ A2, float* __restrict__ s2, float* __restrict__ t2,
    float* __restrict__ s1, float* __restrict__ t1) {
  int i = blockIdx.x * blockDim.x + threadIdx.x;
  if (i < 64) {  // conv2: y = relu(acc*s2 + t2); t2 folds bias+BN
    float s = g2[i] * rsqrtf(v2_[i] + 1e-5f);
    s2[i] = s;
    t2[i] = (b2[i] - m2[i]) * s + be2[i];
  }
  if (i < 32) {  // conv1 likewise
    float s = g1[i] * rsqrtf(v1_[i] + 1e-5f);
    s1[i] = s;
    t1[i] = (b1[i] - m1[i]) * s + be1[i];
  }
  if (i < 4 * 72 * 32) {  // [ct:4][kt:72][lane:32] -> float2 A fragment
    int ct = i / (72 * 32);
    int rem = i % (72 * 32);
    int kt = rem / 32;
    int lane = rem % 32;
    int m = ct * 16 + (lane & 15);
    int k0 = kt * 4 + ((lane >> 4) << 1);
    float2 a;
    {
      int k = k0, ci = k / 9, rr = k % 9;  // rr = ky*3+kx
      a.x = w2[(m * 32 + ci) * 9 + rr];    // w2: [64][32][3][3]
    }
    {
      int k = k0 + 1, ci = k / 9, rr = k % 9;
      a.y = w2[(m * 32 + ci) * 9 + rr];
    }
    A2[i] = a;
  }
}

// ---------------------------------------------------------------- bilateral
__global__ __launch_bounds__(256) void k_bilateral(const float* __restrict__ gray,
                                                   float* __restrict__ smooth) {
  __shared__ float patch[46 * 46];  // (16+2R) x (16+2R) input tile
  __shared__ float wsp[DD * DD];    // spatial Gaussian with circular mask
  const int tx = threadIdx.x, ty = threadIdx.y;
  const int tid = ty * 16 + tx;
  const int gx0 = blockIdx.x * 16, gy0 = blockIdx.y * 16;

  for (int i = tid; i < 46 * 46; i += 256) {
    int py = i / 46, px = i % 46;
    int sy = refl(gy0 + py - RAD, HH);
    int sx = refl(gx0 + px - RAD, WW);
    patch[i] = gray[sy * WW + sx];
  }
  const float inv2ss = 1.0f / (2.0f * 15.0f * 15.0f);
  for (int i = tid; i < DD * DD; i += 256) {
    int dy = i / DD - RAD, dx = i % DD - RAD;
    int r2 = dy * dy + dx * dx;
    wsp[i] = (r2 <= RAD * RAD) ? __expf(-(float)r2 * inv2ss) : 0.0f;
  }
  __syncthreads();

  const float c = patch[(ty + RAD) * 46 + (tx + RAD)];
  const float ninv2sc = -1.0f / (2.0f * 75.0f * 75.0f);
  float num = 0.0f, den = 0.0f;
  for (int ky = 0; ky < DD; ++ky) {
    const float* pr = &patch[(ty + ky) * 46 + tx];
    const float* wr = &wsp[ky * DD];
    for (int kx = 0; kx < DD; ++kx) {
      float p = pr[kx];
      float d = p - c;
      float w = wr[kx] * __expf(d * d * ninv2sc);
      num = fmaf(w, p, num);
      den += w;
    }
  }
  smooth[(gy0 + ty) * WW + (gx0 + tx)] = num / den;
}

// ---------------------------------------------------------------- sobel + max
__global__ __launch_bounds__(256) void k_sobel(const float* __restrict__ sm,
                                               float* __restrict__ mag,
                                               unsigned* __restrict__ maxb) {
  int i = blockIdx.x * blockDim.x + threadIdx.x;
  if (i >= NPIX) return;
  int y = i / WW, x = i % WW;
  float v[3][3];
#pragma unroll
  for (int a = 0; a < 3; ++a)
#pragma unroll
    for (int b = 0; b < 3; ++b)
      v[a][b] = sm[refl(y + a - 1, HH) * WW + refl(x + b - 1, WW)];
  float gx = (v[0][2] - v[0][0]) + 2.0f * (v[1][2] - v[1][0]) + (v[2][2] - v[2][0]);
  float gy = (v[2][0] - v[0][0]) + 2.0f * (v[2][1] - v[0][1]) + (v[2][2] - v[0][2]);
  float m = sqrtf(gx * gx + gy * gy);
  mag[i] = m;
  atomicMax(maxb, __float_as_uint(m));  // m >= 0: uint order == float order
}

// ---------------------------------------------------------------- conv1 (1->32)
__global__ __launch_bounds__(256) void k_conv1(const float* __restrict__ mag,
                                               const unsigned* __restrict__ maxb,
                                               const float* __restrict__ w1,
                                               const float* __restrict__ s1,
                                               const float* __restrict__ t1,
                                               float* __restrict__ f1) {
  __shared__ float wsh[32 * 9];
  __shared__ float ssh[32], tsh[32];
  int tid = threadIdx.x;
  for (int i = tid; i < 288; i += 256) wsh[i] = w1[i];
  if (tid < 32) { ssh[tid] = s1[tid]; tsh[tid] = t1[tid]; }
  __syncthreads();

  int gi = blockIdx.x * blockDim.x + tid;  // 32*NPIX = 3,276,800
  if (gi >= 32 * NPIX) return;
  int c = gi / NPIX;
  int p = gi % NPIX;
  int y = p / WW, x = p % WW;
  float mx = __uint_as_float(*maxb);
  float invm = (mx > 1e-8f) ? (1.0f / mx) : 1.0f;
  float acc = 0.0f;
#pragma unroll
  for (int a = 0; a < 3; ++a) {
    int yy = y + a - 1;
#pragma unroll
    for (int b = 0; b < 3; ++b) {
      int xx = x + b - 1;
      float t = 0.0f;  // zero padding (lax.conv)
      if (yy >= 0 && yy < HH && xx >= 0 && xx < WW) t = mag[yy * WW + xx] * invm;
      acc = fmaf(wsh[c * 9 + a * 3 + b], t, acc);
    }
  }
  float val = fmaf(acc, ssh[c], tsh[c]);
  f1[gi] = fmaxf(val, 0.0f);
}

// -------------------------------------------------- conv2 (32->64) via WMMA f32
// GEMM view: D[64ch][64px] += A[64ch][288k] * B[288k][64px], K-step = 4.
// Block = 128 threads = 4 waves over one row x 64-px strip. Wave w owns
// sub-strip [x0+16w, x0+16w+16) and ALL 64 channels (4 accumulators), so each
// B fragment feeds 4 WMMAs. A fragments stream from the pre-packed global
// buffer (resident in WGP$/L2) with immediate offsets.
// B fragment (4x16 f32, mirror of A): lane l holds column N=l&15,
// K = (l<16 ? 0,1 : 2,3) in VGPR0/VGPR1.
__global__ __launch_bounds__(128) void k_conv2_wmma(const float* __restrict__ f1,
                                                    const v2f* __restrict__ A2,
                                                    const float* __restrict__ s2,
                                                    const float* __restrict__ t2,
                                                    float* __restrict__ f2) {
  __shared__ float lp[32 * 3 * 66];  // [ci][ry 0..2][rx 0..65] im2col patch
  __shared__ int koffB[288];         // k -> byte offset of (ci, ky, kx) in lp
  __shared__ float ssh[64], tsh[64];
  const int tid = threadIdx.x;
  const int bid = blockIdx.x;  // 1600 strips: 320 rows x 5 x-tiles (64 px)
  const int y = bid / 5;
  const int x0 = (bid % 5) * 64;

  for (int i = tid; i < 32 * 3 * 66; i += 128) {
    int ci = i / 198;
    int r = i % 198;
    int ry = r / 66, rx = r % 66;
    int gy = y + ry - 1;
    int gx = x0 + rx - 1;
    float v = 0.0f;  // zero padding
    if (gy >= 0 && gy < HH && gx >= 0 && gx < WW) v = f1[ci * NPIX + gy * WW + gx];
    lp[i] = v;
  }
  for (int k = tid; k < 288; k += 128) {
    int ci = k / 9, rr = k % 9;
    koffB[k] = (ci * 198 + (rr / 3) * 66 + (rr % 3)) * 4;  // bytes
  }
  if (tid < 64) { ssh[tid] = s2[tid]; tsh[tid] = t2[tid]; }
  __syncthreads();

  const int lane = tid & 31;
  const int w = tid >> 5;            // wave id -> 16-px sub-strip
  const int n = lane & 15;           // output pixel within sub-strip
  const int kh = (lane >> 4) << 1;   // 0 or 2 (K pair owned by this lane half)

  const char* lpb = (const char*)lp + (w * 16 + n) * 4;  // per-lane B base
  const int* kp = &koffB[kh];  // kp[4kt], kp[4kt+1]: 8B-aligned pair

  v8f acc0 = {}, acc1 = {}, acc2 = {}, acc3 = {};
#pragma unroll
  for (int kt = 0; kt < 72; ++kt) {
    int o0 = kp[kt * 4];
    int o1 = kp[kt * 4 + 1];
    v2f b;
    b.x = *(const float*)(lpb + o0);
    b.y = *(const float*)(lpb + o1);
    v2f a0 = A2[(0 * 72 + kt) * 32 + lane];
    v2f a1 = A2[(1 * 72 + kt) * 32 + lane];
    v2f a2 = A2[(2 * 72 + kt) * 32 + lane];
    v2f a3 = A2[(3 * 72 + kt) * 32 + lane];
    acc0 = __builtin_amdgcn_wmma_f32_16x16x4_f32(false, a0, false, b, (short)0,
                                                 acc0, false, false);
    acc1 = __builtin_amdgcn_wmma_f32_16x16x4_f32(false, a1, false, b, (short)0,
                                                 acc1, false, false);
    acc2 = __builtin_amdgcn_wmma_f32_16x16x4_f32(false, a2, false, b, (short)0,
                                                 acc2, false, false);
    acc3 = __builtin_amdgcn_wmma_f32_16x16x4_f32(false, a3, false, b, (short)0,
                                                 acc3, false, false);
  }

  // C/D layout: VGPR v -> M = v (lanes 0-15) or v+8 (lanes 16-31); N = lane&15
  const int x = x0 + w * 16 + n;
  const int cb = (lane & 16) ? 8 : 0;
#define STORE_TILE(ACC, CT)                                                 \
  _Pragma("unroll") for (int v = 0; v < 8; ++v) {                           \
    int c = (CT)*16 + cb + v;                                               \
    float val = fmaf((ACC)[v], ssh[c], tsh[c]);                             \
    f2[c * NPIX + y * WW + x] = fmaxf(val, 0.0f);                           \
  }
  STORE_TILE(acc0, 0)
  STORE_TILE(acc1, 1)
  STORE_TILE(acc2, 2)
  STORE_TILE(acc3, 3)
#undef STORE_TILE
}

// ---------------------------------------------------------------- maxpool 2x2
__global__ __launch_bounds__(256) void k_maxpool(const float* __restrict__ f2,
                                                 float* __restrict__ out) {
  int i = blockIdx.x * blockDim.x + threadIdx.x;
  if (i >= 64 * 160 * 160) return;
  int c = i / (160 * 160);
  int r = i % (160 * 160);
  int y = r / 160, x = r % 160;
  const float* p = &f2[c * NPIX + (2 * y) * WW + 2 * x];
  out[i] = fmaxf(fmaxf(p[0], p[1]), fmaxf(p[WW], p[WW + 1]));
}

// ---------------------------------------------------------------- launcher
extern "C" void kernel_launch(void* const* d_in, const int* in_sizes, int n_in,
                              void* d_out, int out_size, void* d_ws, size_t ws_size,
                              hipStream_t stream) {
  (void)in_sizes; (void)n_in; (void)out_size; (void)ws_size;
  const float* img = (const float*)d_in[0];
  const float* w1  = (const float*)d_in[1];
  const float* b1  = (const float*)d_in[2];
  const float* g1  = (const float*)d_in[3];
  const float* be1 = (const float*)d_in[4];
  const float* m1  = (const float*)d_in[5];
  const float* v1  = (const float*)d_in[6];
  const float* w2  = (const float*)d_in[7];
  const float* b2  = (const float*)d_in[8];
  const float* g2  = (const float*)d_in[9];
  const float* be2 = (const float*)d_in[10];
  const float* m2  = (const float*)d_in[11];
  const float* v2  = (const float*)d_in[12];

  float* ws = (float*)d_ws;        // float-element offsets
  float*    gray   = ws;           // 102400
  float*    smooth = ws + 102400;  // 102400
  float*    mag    = ws + 204800;  // 102400
  unsigned* maxb   = (unsigned*)(ws + 307200);
  float*    s1     = ws + 307232;  // 32
  float*    t1     = ws + 307264;  // 32
  float*    s2     = ws + 307296;  // 64
  float*    t2     = ws + 307360;  // 64
  float2*   A2     = (float2*)(ws + 307456);  // 9216 float2 (8B aligned)
  float*    f1     = ws + 325888;  // 32*102400
  float*    f2     = ws + 3602688; // 64*102400  (total ~40.6 MB)
  float*    out    = (float*)d_out;  // 64*160*160

  k_gray<<<400, 256, 0, stream>>>(img, gray, maxb);
  k_prep<<<36, 256, 0, stream>>>(w2, b2, g2, be2, m2, v2, b1, g1, be1, m1, v1,
                                 A2, s2, t2, s1, t1);
  k_bilateral<<<dim3(20, 20), dim3(16, 16), 0, stream>>>(gray, smooth);
  k_sobel<<<400, 256, 0, stream>>>(smooth, mag, maxb);
  k_conv1<<<12800, 256, 0, stream>>>(mag, maxb, w1, s1, t1, f1);
  k_conv2_wmma<<<1600, 128, 0, stream>>>(f1, (const v2f*)A2, s2, t2, f2);
  k_maxpool<<<6400, 256, 0, stream>>>(f2, out);
}